// GraphConv_41815801594346
// MI455X (gfx1250) — compile-verified
//
#include <hip/hip_runtime.h>

typedef __attribute__((ext_vector_type(2))) float v2f;
typedef __attribute__((ext_vector_type(4))) float v4f;
typedef __attribute__((ext_vector_type(8))) float v8f;

typedef unsigned int u32x4 __attribute__((ext_vector_type(4)));
typedef int          i32x4 __attribute__((ext_vector_type(4)));
typedef int          i32x8 __attribute__((ext_vector_type(8)));

#define BN 2
#define VN 50000
#define CN 128
#define FN 128
#define TILE_M 128   // rows per workgroup (8 waves x 16 rows)

// ---------------------------------------------------------------------------
// Kernel 0: zero the output (harness poisons d_out with 0xAA)
// ---------------------------------------------------------------------------
__global__ __launch_bounds__(256) void zero_f32(float* __restrict__ p, int n) {
  int i = blockIdx.x * blockDim.x + threadIdx.x;       // one float4 per thread
  if (i * 4 < n) {
    v4f z = {0.f, 0.f, 0.f, 0.f};
    ((v4f*)p)[i] = z;
  }
}

// ---------------------------------------------------------------------------
// Kernel 1: h = x @ W + b  via V_WMMA_F32_16X16X4_F32
//   grid = (ceil(V/128), B), block = 256 (8 waves)
//   wave w owns output rows [tile*128 + w*16, +16) x all 128 cols.
//   W (64 KB) staged to LDS: TDM tensor_load_to_lds (when available) in
//   parallel with a manual float4 staging loop writing identical bytes.
//   A fragments are double-buffered in registers to hide global load latency.
// ---------------------------------------------------------------------------
__global__ __launch_bounds__(256) void gemm_bias(const float* __restrict__ x,
                                                 const float* __restrict__ W,
                                                 const float* __restrict__ bias,
                                                 float* __restrict__ h) {
  __shared__ float Wlds[CN * FN];                      // 64 KB, row-major (k, f)

  const int b    = blockIdx.y;
  const int tid  = threadIdx.x;
  const int wave = tid >> 5;
  const int lane = tid & 31;
  const int half = lane >> 4;                          // 0: lanes 0-15, 1: lanes 16-31
  const int nl   = lane & 15;

#if defined(__gfx1250__) && __has_builtin(__builtin_amdgcn_tensor_load_to_lds)
  // Tensor Data Mover: DMA the 128x128 f32 W tile into LDS.
  if (tid == 0) {
    const unsigned long long ga = (unsigned long long)(const void*)W;
    const unsigned int lds_addr = (unsigned int)(unsigned long long)(const void*)Wlds;
    // D# group0: count=1 | lds_addr | global_addr[56:0] | type=2
    u32x4 g0 = { 1u, lds_addr, (unsigned int)ga,
                 (unsigned int)((ga >> 32) & 0x1FFFFFFu) | 0x80000000u };
    // D# group1: data_size=4B; tensor 128x128; tile 128x128; strides 128
    i32x8 g1 = { 0x20000,            // data_size=2 (4B), mask=0, no pad/iter
                 (CN << 16),         // tensor_dim0 = 128 (bits 79:48 low half)
                 (CN << 16),         // tensor_dim1 = 128 (bits 111:80 low half)
                 (CN << 16),         // tile_dim0   = 128 (bits 127:112)
                 CN,                 // tile_dim1   = 128, tile_dim2 = 0
                 CN,                 // tensor_dim0_stride = 128 (low 32)
                 (CN << 16),         // tensor_dim1_stride = 128 (bits 255:208 low)
                 0 };
    i32x4 gz4 = { 0, 0, 0, 0 };      // 2D tensor: groups 2/3 unused
#if __clang_major__ >= 23
    i32x8 gz8 = { 0, 0, 0, 0, 0, 0, 0, 0 };
    __builtin_amdgcn_tensor_load_to_lds(g0, g1, gz4, gz4, gz8, 0);
#else
    __builtin_amdgcn_tensor_load_to_lds(g0, g1, gz4, gz4, 0);
#endif
  }
  __builtin_amdgcn_s_wait_tensorcnt(0);
#endif

  // Manual staging (same bytes as the TDM copy keeps this correct regardless)
  for (int i = tid; i < (CN * FN) / 4; i += 256) {
    ((v4f*)Wlds)[i] = ((const v4f*)W)[i];
  }
  __syncthreads();

  // A-fragment row this lane supplies. Clamp (not zero-pad) tail rows: the
  // store is guarded, so garbage accumulators for rows >= VN are harmless,
  // and the load stays unconditional -> no divergent branch, pipelinable.
  const int arow  = blockIdx.x * TILE_M + wave * 16 + nl;
  const int asafe = arow < VN ? arow : VN - 1;
  const float* xrow = x + (size_t)b * VN * CN + (size_t)asafe * CN + 2 * half;

  v8f acc[8];
  #pragma unroll
  for (int n = 0; n < 8; ++n) {
    v8f z = {0.f, 0.f, 0.f, 0.f, 0.f, 0.f, 0.f, 0.f};
    acc[n] = z;
  }

  // K loop: each WMMA consumes K=4. A layout (ISA 7.12.2): lanes 0-15 carry
  // K=k,k+1; lanes 16-31 carry K=k+2,k+3. B mirrors the K striping, N=lane&15.
  // A is double-buffered: the load for k+4 issues before the WMMA batch for k,
  // so its latency hides under 8 WMMAs + 8 LDS loads. Tail wraps to k=0 (safe,
  // L0-hot, result discarded).
  v2f a = *(const v2f*)(xrow);                         // k = 0
  for (int k = 0; k < CN; k += 4) {
    const int knext = (k + 4 < CN) ? (k + 4) : 0;
    v2f anext = *(const v2f*)(xrow + knext);           // aligned global_load_b64
    #pragma unroll
    for (int n = 0; n < 8; ++n) {
      v2f bb;
      bb[0] = Wlds[(k + 2 * half)     * FN + n * 16 + nl];
      bb[1] = Wlds[(k + 2 * half + 1) * FN + n * 16 + nl];
      acc[n] = __builtin_amdgcn_wmma_f32_16x16x4_f32(
          /*neg_a=*/false, a, /*neg_b=*/false, bb,
          /*c_mod=*/(short)0, acc[n], /*reuse_a=*/false, /*reuse_b=*/false);
    }
    a = anext;
  }

  // D layout: element r of v8f is (M = r + 8*half, N = lane&15)
  float* hb = h + (size_t)b * VN * FN;
  #pragma unroll
  for (int r = 0; r < 8; ++r) {
    const int m = blockIdx.x * TILE_M + wave * 16 + half * 8 + r;
    if (m < VN) {
      #pragma unroll
      for (int n = 0; n < 8; ++n) {
        const int col = n * 16 + nl;
        hb[(size_t)m * FN + col] = acc[n][r] + bias[col];
      }
    }
  }
}

// ---------------------------------------------------------------------------
// Kernel 2: out[b,r,:] += val * h[b,c,:]  (one wave per edge, 4 floats/lane)
// ---------------------------------------------------------------------------
__global__ __launch_bounds__(256) void spmm_scatter(const float* __restrict__ h,
                                                    const int* __restrict__ rows,
                                                    const int* __restrict__ cols,
                                                    const float* __restrict__ vals,
                                                    float* __restrict__ out, int E) {
  const int e    = (blockIdx.x * blockDim.x + threadIdx.x) >> 5;
  const int lane = threadIdx.x & 31;
  if (e >= E) return;

  const int   r = rows[e];
  const int   c = cols[e];
  const float v = vals[e];

  #pragma unroll
  for (int b = 0; b < BN; ++b) {
    const float* hp = h   + ((size_t)b * VN + c) * FN + lane * 4;
    float*       op = out + ((size_t)b * VN + r) * FN + lane * 4;
    v4f hv = *(const v4f*)hp;
    atomicAdd(&op[0], v * hv[0]);
    atomicAdd(&op[1], v * hv[1]);
    atomicAdd(&op[2], v * hv[2]);
    atomicAdd(&op[3], v * hv[3]);
  }
}

// ---------------------------------------------------------------------------
extern "C" void kernel_launch(void* const* d_in, const int* in_sizes, int n_in,
                              void* d_out, int out_size, void* d_ws, size_t ws_size,
                              hipStream_t stream) {
  const float* x        = (const float*)d_in[0];
  const int*   adj_rows = (const int*)d_in[1];
  const int*   adj_cols = (const int*)d_in[2];
  const float* adj_vals = (const float*)d_in[3];
  const float* W        = (const float*)d_in[4];
  const float* bias     = (const float*)d_in[5];
  float*       out      = (float*)d_out;
  float*       h        = (float*)d_ws;        // B*V*F floats = 51.2 MB scratch
  const int    E        = in_sizes[1];

  // 1) zero output
  const int total = BN * VN * FN;
  zero_f32<<<(total / 4 + 255) / 256, 256, 0, stream>>>(out, total);

  // 2) dense projection via fp32 WMMA (+ TDM-staged W)
  dim3 g((VN + TILE_M - 1) / TILE_M, BN);
  gemm_bias<<<g, 256, 0, stream>>>(x, W, bias, h);

  // 3) sparse scatter-add (one wave per edge)
  const long long threads = (long long)E * 32;
  spmm_scatter<<<(int)((threads + 255) / 256), 256, 0, stream>>>(
      h, adj_rows, adj_cols, adj_vals, out, E);
}